// GCN_83073257439789
// MI455X (gfx1250) — compile-verified
//
#include <hip/hip_runtime.h>

typedef __attribute__((ext_vector_type(2))) float v2f;
typedef __attribute__((ext_vector_type(8))) float v8f;

#define F_IN  32
#define HC    64
#define C_OUT 8

// ---------------- degree / normalization ----------------

__global__ __launch_bounds__(256) void k_deg_init(float* __restrict__ deg, int N) {
  int i = blockIdx.x * 256 + threadIdx.x;
  if (i < N) deg[i] = 1.0f;                      // self-loop contributes 1
}

__global__ __launch_bounds__(256) void k_deg_count(const int* __restrict__ dst,
                                                   float* __restrict__ deg, int E) {
  int e = blockIdx.x * 256 + threadIdx.x;
  if (e < E) atomicAdd(&deg[dst[e]], 1.0f);      // no-return global_atomic_add_f32
}

__global__ __launch_bounds__(256) void k_dinv(float* __restrict__ deg, int N) {
  int i = blockIdx.x * 256 + threadIdx.x;
  if (i < N) deg[i] = 1.0f / sqrtf(deg[i]);      // deg >= 1 always
}

// ---------------- GEMM 1: h1[N,64] = x[N,32] @ W1[32,64]  (v_wmma_f32_16x16x4_f32) ----------------
// One wave per 16-row tile; 8 K-steps x 4 N-tiles = 32 WMMAs.
// A f32 16x4 layout: lanes 0-15 = rows, VGPR0/1 = K,K+1 (lanes<16) or K+2,K+3 (lanes>=16).
// C/D layout: VGPR r -> M=r (lanes 0-15) / M=8+r (lanes 16-31), N = lane&15.

__global__ __launch_bounds__(32) void k_gemm1(const float* __restrict__ x,
                                              const float* __restrict__ W1,
                                              float* __restrict__ h1, int N) {
  const int lane  = threadIdx.x;
  const int l15   = lane & 15;
  const int khalf = lane >> 4;
  int row = blockIdx.x * 16 + l15;
  if (row >= N) row = N - 1;                     // clamp loads; EXEC stays all-1 for WMMA
  v8f c[4] = {};
  #pragma unroll
  for (int k0 = 0; k0 < F_IN; k0 += 4) {
    const int ka = k0 + khalf * 2;
    v2f a = { x[(size_t)row * F_IN + ka], x[(size_t)row * F_IN + ka + 1] };
    #pragma unroll
    for (int nt = 0; nt < 4; ++nt) {
      const int col = nt * 16 + l15;
      v2f b = { W1[(size_t)ka * HC + col], W1[(size_t)(ka + 1) * HC + col] };
      c[nt] = __builtin_amdgcn_wmma_f32_16x16x4_f32(false, a, false, b,
                                                    (short)0, c[nt], false, false);
    }
  }
  const int mbase = blockIdx.x * 16 + khalf * 8;
  #pragma unroll
  for (int nt = 0; nt < 4; ++nt)
    #pragma unroll
    for (int r = 0; r < 8; ++r) {
      const int m = mbase + r;
      if (m < N) h1[(size_t)m * HC + nt * 16 + l15] = c[nt][r];
    }
}

// ---------------- layer-1 aggregation ----------------
// out1[i] = b1 + h1[i]*dinv[i]^2  (bias + self-loop), then scatter-add edges.

__global__ __launch_bounds__(256) void k_agg1_init(const float* __restrict__ h1,
                                                   const float* __restrict__ b1,
                                                   const float* __restrict__ dinv,
                                                   float* __restrict__ out1, int N) {
  int i = blockIdx.x * 256 + threadIdx.x;        // one float4 (16 per node)
  if (i >= N * (HC / 4)) return;
  const int node = i >> 4;
  const int fq   = i & 15;
  const float di = dinv[node];
  const float s  = di * di;
  float4 v  = ((const float4*)h1)[i];
  float4 bb = ((const float4*)b1)[fq];
  float4 o;
  o.x = bb.x + v.x * s;  o.y = bb.y + v.y * s;
  o.z = bb.z + v.z * s;  o.w = bb.w + v.w * s;
  ((float4*)out1)[i] = o;
}

__global__ __launch_bounds__(256) void k_agg1_edges(const int* __restrict__ src,
                                                    const int* __restrict__ dst,
                                                    const float* __restrict__ dinv,
                                                    const float* __restrict__ h1,
                                                    float* __restrict__ out1, int E) {
  unsigned gid = blockIdx.x * 256u + threadIdx.x;
  const int e  = gid >> 4;                       // 16 lanes share one edge
  const int fq = gid & 15;                       // each lane handles a float4
  if (e >= E) return;
  const int s = src[e], d = dst[e];
  const float nrm = dinv[s] * dinv[d];
  float4 v = ((const float4*)(h1 + (size_t)s * HC))[fq];
  float* o = out1 + (size_t)d * HC + fq * 4;
  atomicAdd(o + 0, v.x * nrm);
  atomicAdd(o + 1, v.y * nrm);
  atomicAdd(o + 2, v.z * nrm);
  atomicAdd(o + 3, v.w * nrm);
}

// ---------------- GEMM 2: h2[N,8] = out1[N,64] @ W2[64,8]  (N-dim padded to 16) ----------------

__global__ __launch_bounds__(32) void k_gemm2(const float* __restrict__ h,
                                              const float* __restrict__ W2,
                                              float* __restrict__ h2, int N) {
  const int lane  = threadIdx.x;
  const int l15   = lane & 15;
  const int khalf = lane >> 4;
  int row = blockIdx.x * 16 + l15;
  if (row >= N) row = N - 1;
  const bool live_col = (l15 < C_OUT);
  v8f c = {};
  #pragma unroll
  for (int k0 = 0; k0 < HC; k0 += 4) {
    const int ka = k0 + khalf * 2;
    v2f a = { h[(size_t)row * HC + ka], h[(size_t)row * HC + ka + 1] };
    v2f b = { live_col ? W2[(size_t)ka * C_OUT + l15] : 0.0f,
              live_col ? W2[(size_t)(ka + 1) * C_OUT + l15] : 0.0f };
    c = __builtin_amdgcn_wmma_f32_16x16x4_f32(false, a, false, b,
                                              (short)0, c, false, false);
  }
  const int mbase = blockIdx.x * 16 + khalf * 8;
  if (live_col) {
    #pragma unroll
    for (int r = 0; r < 8; ++r) {
      const int m = mbase + r;
      if (m < N) h2[(size_t)m * C_OUT + l15] = c[r];
    }
  }
}

// ---------------- layer-2 aggregation ----------------

__global__ __launch_bounds__(256) void k_out_init(const float* __restrict__ h2,
                                                  const float* __restrict__ b2,
                                                  const float* __restrict__ dinv,
                                                  float* __restrict__ out, int N) {
  int i = blockIdx.x * 256 + threadIdx.x;        // one float4 (2 per node)
  if (i >= N * (C_OUT / 4)) return;
  const int node = i >> 1;
  const int half = i & 1;
  const float di = dinv[node];
  const float s  = di * di;
  float4 v  = ((const float4*)h2)[i];
  float4 bb = ((const float4*)b2)[half];
  float4 o;
  o.x = bb.x + v.x * s;  o.y = bb.y + v.y * s;
  o.z = bb.z + v.z * s;  o.w = bb.w + v.w * s;
  ((float4*)out)[i] = o;
}

__global__ __launch_bounds__(256) void k_agg2_edges(const int* __restrict__ src,
                                                    const int* __restrict__ dst,
                                                    const float* __restrict__ dinv,
                                                    const float* __restrict__ h2,
                                                    float* __restrict__ out, int E) {
  unsigned gid = blockIdx.x * 256u + threadIdx.x;
  const int e    = gid >> 1;                     // 2 lanes per edge
  const int half = gid & 1;
  if (e >= E) return;
  const int s = src[e], d = dst[e];
  const float nrm = dinv[s] * dinv[d];
  float4 v = ((const float4*)(h2 + (size_t)s * C_OUT))[half];
  float* o = out + (size_t)d * C_OUT + half * 4;
  atomicAdd(o + 0, v.x * nrm);
  atomicAdd(o + 1, v.y * nrm);
  atomicAdd(o + 2, v.z * nrm);
  atomicAdd(o + 3, v.w * nrm);
}

// ---------------- launcher ----------------

extern "C" void kernel_launch(void* const* d_in, const int* in_sizes, int n_in,
                              void* d_out, int out_size, void* d_ws, size_t ws_size,
                              hipStream_t stream) {
  const float* x  = (const float*)d_in[0];
  const int*   ei = (const int*)d_in[1];        // JAX x64 off -> int32
  const float* W1 = (const float*)d_in[2];
  const float* b1 = (const float*)d_in[3];
  const float* W2 = (const float*)d_in[4];
  const float* b2 = (const float*)d_in[5];
  float* out = (float*)d_out;

  const int N = in_sizes[0] / F_IN;
  const int E = in_sizes[1] / 2;
  const int* src = ei;
  const int* dst = ei + E;

  // workspace: dinv[N] | h1[N*64] | out1[N*64] | h2[N*8]   (~55 MB)
  float* dinv = (float*)d_ws;
  float* h1   = dinv + N;
  float* out1 = h1 + (size_t)N * HC;
  float* h2   = out1 + (size_t)N * HC;

  const int nbN  = (N + 255) / 256;
  const int nbE  = (E + 255) / 256;
  const int tiles = (N + 15) / 16;

  k_deg_init <<<nbN, 256, 0, stream>>>(dinv, N);
  k_deg_count<<<nbE, 256, 0, stream>>>(dst, dinv, E);
  k_dinv     <<<nbN, 256, 0, stream>>>(dinv, N);

  k_gemm1<<<tiles, 32, 0, stream>>>(x, W1, h1, N);

  k_agg1_init <<<(N * (HC / 4) + 255) / 256, 256, 0, stream>>>(h1, b1, dinv, out1, N);
  k_agg1_edges<<<(unsigned)(((size_t)E * 16 + 255) / 256), 256, 0, stream>>>(src, dst, dinv, h1, out1, E);

  k_gemm2<<<tiles, 32, 0, stream>>>(out1, W2, h2, N);

  k_out_init  <<<(N * (C_OUT / 4) + 255) / 256, 256, 0, stream>>>(h2, b2, dinv, out, N);
  k_agg2_edges<<<(unsigned)(((size_t)E * 2 + 255) / 256), 256, 0, stream>>>(src, dst, dinv, h2, out, E);
}